// NodeEmbedding_70317204570425
// MI455X (gfx1250) — compile-verified
//
#include <hip/hip_runtime.h>

#define N_NODES 100000
#define N_EDGES 1600000
#define DIM 128
#define DEG_PAD 100096  // N_NODES padded to multiple of 64 floats

typedef float v2f __attribute__((ext_vector_type(2)));
typedef float v8f __attribute__((ext_vector_type(8)));

// ---------------- degree accumulation ----------------

__global__ void k_init_deg(float* __restrict__ deg) {
  int i = blockIdx.x * blockDim.x + threadIdx.x;
  if (i < N_NODES) deg[i] = 0.0f;
}

__global__ void k_deg_accum(const long long* __restrict__ ei,
                            const float* __restrict__ ew,
                            float* __restrict__ deg) {
  int e = blockIdx.x * blockDim.x + threadIdx.x;
  if (e < N_EDGES) {
    int r = (int)ei[e];  // row = edge_index[0][e]
    __hip_atomic_fetch_add(&deg[r], ew[e], __ATOMIC_RELAXED,
                           __HIP_MEMORY_SCOPE_AGENT);
  }
}

__global__ void k_dinv(float* __restrict__ deg) {
  int i = blockIdx.x * blockDim.x + threadIdx.x;
  // deg+1 >= 1 always (self loop), so rsqrt is safe; overwrite in place.
  if (i < N_NODES) deg[i] = __frsqrt_rn(deg[i] + 1.0f);
}

// ---------------- dense projection h = X @ W via WMMA f32 ----------------
// One wave -> one 16x16 tile of h. 8 waves/block cover all 128 columns.
// A 16x4 layout: lane<16: (M=lane, K=k0+{0,1}); lane>=16: (M=lane-16, K=k0+{2,3})
// B 4x16 layout: v0 holds rows k0+{0,2}, v1 holds rows k0+{1,3}, N=lane&15
// C/D: VGPR r holds (M = r + 8*(lane>>4), N = lane&15)

__global__ void __launch_bounds__(256)
k_gemm_wmma(const float* __restrict__ X, const float* __restrict__ W,
            float* __restrict__ H) {
  const int lane = threadIdx.x & 31;
  const int wave = threadIdx.x >> 5;
  const int half = lane >> 4;
  const int ml   = lane & 15;
  const int m0   = blockIdx.x * 16;
  const int n0   = wave * 16;

  const float* __restrict__ arow = X + (size_t)(m0 + ml) * DIM + 2 * half;
  const float* __restrict__ bcol = W + (size_t)(2 * half) * DIM + n0 + ml;

  v8f c = {};
#pragma unroll
  for (int k0 = 0; k0 < DIM; k0 += 4) {
    v2f a = *(const v2f*)(arow + k0);          // X[m][k0+2h .. k0+2h+1]
    const float* bp = bcol + (size_t)k0 * DIM; // W[k0+2h][n], W[k0+2h+1][n]
    v2f b;
    b.x = bp[0];
    b.y = bp[DIM];
    c = __builtin_amdgcn_wmma_f32_16x16x4_f32(
        /*neg_a=*/false, a, /*neg_b=*/false, b,
        /*c_mod=*/(short)0, c, /*reuse_a=*/false, /*reuse_b=*/false);
  }

  float* __restrict__ hp = H + (size_t)(m0 + 8 * half) * DIM + n0 + ml;
#pragma unroll
  for (int r = 0; r < 8; ++r) hp[(size_t)r * DIM] = c[r];
}

// ---------------- self-loop term: acc = dinv^2 * h ----------------

__global__ void k_selfloop(const float* __restrict__ h,
                           const float* __restrict__ dinv,
                           float* __restrict__ acc) {
  size_t i = (size_t)blockIdx.x * blockDim.x + threadIdx.x;
  if (i < (size_t)N_NODES * DIM) {
    int node = (int)(i >> 7);
    float d = dinv[node];
    acc[i] = d * d * h[i];
  }
}

// ---------------- SpMM edge scatter: acc[row] += norm * h[col] ----------------
// One wave per edge; each lane handles 4 consecutive columns (float4 gather,
// 4 fp32 atomics). h + acc both live in the 192MB L2, so this stays on-chip.

__global__ void __launch_bounds__(256)
k_spmm(const long long* __restrict__ ei, const float* __restrict__ ew,
       const float* __restrict__ dinv, const float* __restrict__ h,
       float* __restrict__ acc) {
  const int lane = threadIdx.x & 31;
  const int e = blockIdx.x * (256 / 32) + (threadIdx.x >> 5);
  if (e >= N_EDGES) return;

  const int r = (int)ei[e];            // destination row
  const int cl = (int)ei[N_EDGES + e]; // source col
  const float w = dinv[r] * ew[e] * dinv[cl];

  const float4* __restrict__ hp = (const float4*)(h + (size_t)cl * DIM);
  float* __restrict__ ap = acc + (size_t)r * DIM + lane * 4;
  float4 v = hp[lane];
  __hip_atomic_fetch_add(&ap[0], w * v.x, __ATOMIC_RELAXED, __HIP_MEMORY_SCOPE_AGENT);
  __hip_atomic_fetch_add(&ap[1], w * v.y, __ATOMIC_RELAXED, __HIP_MEMORY_SCOPE_AGENT);
  __hip_atomic_fetch_add(&ap[2], w * v.z, __ATOMIC_RELAXED, __HIP_MEMORY_SCOPE_AGENT);
  __hip_atomic_fetch_add(&ap[3], w * v.w, __ATOMIC_RELAXED, __HIP_MEMORY_SCOPE_AGENT);
}

// ---------------- bias + relu ----------------

__global__ void k_final(const float* __restrict__ acc,
                        const float* __restrict__ bias,
                        float* __restrict__ out) {
  size_t i = (size_t)blockIdx.x * blockDim.x + threadIdx.x;
  if (i < (size_t)N_NODES * DIM) {
    float v = acc[i] + bias[i & 127];
    out[i] = v > 0.0f ? v : 0.0f;
  }
}

// ---------------- launcher ----------------

extern "C" void kernel_launch(void* const* d_in, const int* in_sizes, int n_in,
                              void* d_out, int out_size, void* d_ws, size_t ws_size,
                              hipStream_t stream) {
  const float*     x    = (const float*)d_in[0];      // [N,128]
  const long long* ei   = (const long long*)d_in[1];  // [2,E] int64
  const float*     ew   = (const float*)d_in[2];      // [E]
  const float*     w    = (const float*)d_in[3];      // [128,128]
  const float*     bias = (const float*)d_in[4];      // [128]
  float*           out  = (float*)d_out;              // [N,128]

  float* deg = (float*)d_ws;                          // N floats (becomes dinv)
  float* h   = deg + DEG_PAD;                         // N*128 floats
  float* acc = h + (size_t)N_NODES * DIM;             // N*128 floats

  k_init_deg <<<(N_NODES + 255) / 256, 256, 0, stream>>>(deg);
  k_deg_accum<<<(N_EDGES + 255) / 256, 256, 0, stream>>>(ei, ew, deg);
  k_dinv     <<<(N_NODES + 255) / 256, 256, 0, stream>>>(deg);
  k_gemm_wmma<<<N_NODES / 16, 256, 0, stream>>>(x, w, h);
  k_selfloop <<<(N_NODES * DIM + 255) / 256, 256, 0, stream>>>(h, deg, acc);
  k_spmm     <<<(N_EDGES + 7) / 8, 256, 0, stream>>>(ei, ew, deg, h, acc);
  k_final    <<<(N_NODES * DIM + 255) / 256, 256, 0, stream>>>(acc, bias, out);
}